// GPTAttention_32100585570934
// MI455X (gfx1250) — compile-verified
//
#include <hip/hip_runtime.h>
#include <math.h>

typedef float v2f __attribute__((ext_vector_type(2)));
typedef float v8f __attribute__((ext_vector_type(8)));

// Problem constants (reference: B=2, N=2048, E=2048, H=16, D=128)
constexpr int NB   = 2;
constexpr int SEQ  = 2048;
constexpr int EMB  = 2048;
constexpr int NH   = 16;
constexpr int HD   = 128;
constexpr int E3   = 3 * EMB;          // 6144
constexpr int ROWS = NB * SEQ;         // 4096
constexpr int BH   = NB * NH;          // 32
constexpr int NT16 = SEQ / 16;         // 128 row-tiles per head

static __device__ __forceinline__ v8f wmma_f32(v2f a, v2f b, v8f c) {
  // V_WMMA_F32_16X16X4_F32 : D = A(16x4,f32) * B(4x16,f32) + C(16x16,f32)
  return __builtin_amdgcn_wmma_f32_16x16x4_f32(false, a, false, b, (short)0, c,
                                               false, false);
}

// exchange value between lane l and lane l^16 (SWAPX16 via ds_swizzle)
static __device__ __forceinline__ float swap16(float x) {
  int i = __builtin_amdgcn_ds_swizzle(__float_as_int(x), 0x401F);
  return __int_as_float(i);
}

// ---------------------------------------------------------------------------
// Stage 1: QKV = X[4096,2048] @ Wf[2048,6144], scattered to q/k/v [B,H,N,D]
// one wave per 32x64 macro tile (2x4 grid of 16x16 WMMA tiles)
// ---------------------------------------------------------------------------
__global__ __launch_bounds__(32) void qkv_gemm_kernel(
    const float* __restrict__ X, const float* __restrict__ W,
    float* __restrict__ q, float* __restrict__ k, float* __restrict__ v) {
  const int lane = threadIdx.x & 31;
  const int half = lane >> 4;    // 0: K0/K1 (or M+0), 1: K2/K3 (or M+8)
  const int mn   = lane & 15;    // A: row M, B: col N
  const int row0 = blockIdx.y * 32;
  const int col0 = blockIdx.x * 64;

  v8f acc[2][4];
  v8f z = {};
  #pragma unroll
  for (int i = 0; i < 2; ++i)
    #pragma unroll
    for (int t = 0; t < 4; ++t) acc[i][t] = z;

  const float* Xr0 = X + (size_t)(row0 + mn) * EMB;
  const float* Xr1 = X + (size_t)(row0 + 16 + mn) * EMB;

  for (int c = 0; c < EMB / 4; ++c) {
    const int kb = 4 * c + 2 * half;
    v2f a0, a1;
    a0.x = Xr0[kb]; a0.y = Xr0[kb + 1];
    a1.x = Xr1[kb]; a1.y = Xr1[kb + 1];
    const float* Wk0 = W + (size_t)kb * E3 + col0 + mn;
    const float* Wk1 = Wk0 + E3;
    #pragma unroll
    for (int t = 0; t < 4; ++t) {
      v2f b; b.x = Wk0[16 * t]; b.y = Wk1[16 * t];
      acc[0][t] = wmma_f32(a0, b, acc[0][t]);
      acc[1][t] = wmma_f32(a1, b, acc[1][t]);
    }
  }

  #pragma unroll
  for (int mi = 0; mi < 2; ++mi)
    #pragma unroll
    for (int t = 0; t < 4; ++t)
      #pragma unroll
      for (int r = 0; r < 8; ++r) {
        const int gm = row0 + mi * 16 + r + 8 * half;  // 0..4095 (b*N+n)
        const int gc = col0 + 16 * t + mn;             // 0..6143
        const int which = gc >> 11;                    // 0:q 1:k 2:v
        const int e  = gc & (EMB - 1);
        const int h  = e >> 7, d = e & (HD - 1);
        const int bb = gm >> 11, n = gm & (SEQ - 1);
        float* dst = (which == 0) ? q : (which == 1) ? k : v;
        dst[((size_t)(bb * NH + h) * SEQ + n) * HD + d] = acc[mi][t][r];
      }
}

// ---------------------------------------------------------------------------
// Stage 2: causal scores + softmax. One wave per (b,h,16-row strip).
// Q fragments held in registers; 16xSEQ score strip staged in 128KB LDS.
// ---------------------------------------------------------------------------
__global__ __launch_bounds__(32) void attn_softmax_kernel(
    const float* __restrict__ q, const float* __restrict__ kk,
    float* __restrict__ attn /* [B,H,N,N] */) {
  const int lane = threadIdx.x & 31;
  const int half = lane >> 4;
  const int mn   = lane & 15;
  const int s  = blockIdx.x;
  const int it = s & (NT16 - 1);
  const int bh = s >> 7;
  const int i0 = it * 16;

  const float* Q  = q  + (size_t)bh * SEQ * HD;
  const float* Kh = kk + (size_t)bh * SEQ * HD;
  extern __shared__ float sS[];  // [16][SEQ]

  // preload Q fragments for all 32 K-chunks (64 VGPRs)
  v2f qa[HD / 4];
  const float* Qr = Q + (size_t)(i0 + mn) * HD;
  #pragma unroll
  for (int c = 0; c < HD / 4; ++c) {
    const int kb = 4 * c + 2 * half;
    qa[c].x = Qr[kb]; qa[c].y = Qr[kb + 1];
  }

  const float scale = 0.08838834764831845f;  // 1/sqrt(128)
  const float MASKV = -3.402823466e+38f;     // finfo(f32).min

  for (int jt = 0; jt <= it; ++jt) {
    const int j0 = jt * 16;
    v8f acc = {};
    const float* Kr = Kh + (size_t)(j0 + mn) * HD;  // B[k][n] = K[j0+n][k]
    #pragma unroll
    for (int c = 0; c < HD / 4; ++c) {
      const int kb = 4 * c + 2 * half;
      v2f b; b.x = Kr[kb]; b.y = Kr[kb + 1];
      acc = wmma_f32(qa[c], b, acc);
    }
    #pragma unroll
    for (int r = 0; r < 8; ++r) {
      const int i = i0 + r + 8 * half;
      const int j = j0 + mn;
      float val = acc[r] * scale;
      if (j > i) val = MASKV;
      sS[(r + 8 * half) * SEQ + j] = val;
    }
  }
  __syncthreads();

  const int limit = (it + 1) * 16;  // valid columns [0, limit)
  {
    // lanes (l, l+16) cooperate on row l&15
    float* rowp = sS + mn * SEQ;
    float mx = -3.402823466e+38f;
    for (int j = half; j < limit; j += 2) mx = fmaxf(mx, rowp[j]);
    mx = fmaxf(mx, swap16(mx));
    float sum = 0.0f;
    for (int j = half; j < limit; j += 2) {
      const float ev = __expf(rowp[j] - mx);
      rowp[j] = ev;
      sum += ev;
    }
    sum += swap16(sum);
    const float inv = 1.0f / sum;
    for (int j = half; j < limit; j += 2) rowp[j] *= inv;
  }
  __syncthreads();

  // coalesced write of full 16xSEQ strip (zeros beyond causal limit)
  float* dst = attn + (size_t)bh * SEQ * SEQ + (size_t)i0 * SEQ;
  for (int idx = lane; idx < 16 * SEQ; idx += 32) {
    const int row = idx >> 11;
    const int j   = idx & (SEQ - 1);
    dst[(size_t)row * SEQ + j] = (j < limit) ? sS[row * SEQ + j] : 0.0f;
  }
}

// ---------------------------------------------------------------------------
// Stage 3: attn_out = P @ V per head, causal-trimmed K loop.
// One wave per 16x128 output strip; writes [B,N,h*D+d] layout for proj GEMM.
// ---------------------------------------------------------------------------
__global__ __launch_bounds__(32) void pv_gemm_kernel(
    const float* __restrict__ attn, const float* __restrict__ v,
    float* __restrict__ ao /* [B,N,E] */) {
  const int lane = threadIdx.x & 31;
  const int half = lane >> 4;
  const int mn   = lane & 15;
  const int s  = blockIdx.x;
  const int it = s & (NT16 - 1);
  const int bh = s >> 7;
  const int i0 = it * 16;

  const float* P  = attn + (size_t)bh * SEQ * SEQ;
  const float* Vh = v + (size_t)bh * SEQ * HD;

  v8f acc[8];
  v8f z = {};
  #pragma unroll
  for (int t = 0; t < 8; ++t) acc[t] = z;

  const int limit = (it + 1) * 16;  // P row i0.. has zeros beyond this
  const float* Pr = P + (size_t)(i0 + mn) * SEQ;
  for (int c = 0; c < limit / 4; ++c) {
    const int kb = 4 * c + 2 * half;
    v2f a; a.x = Pr[kb]; a.y = Pr[kb + 1];
    const float* Vk0 = Vh + (size_t)kb * HD + mn;
    const float* Vk1 = Vk0 + HD;
    #pragma unroll
    for (int t = 0; t < 8; ++t) {
      v2f b; b.x = Vk0[16 * t]; b.y = Vk1[16 * t];
      acc[t] = wmma_f32(a, b, acc[t]);
    }
  }

  const int b = bh >> 4, h = bh & (NH - 1);
  #pragma unroll
  for (int t = 0; t < 8; ++t)
    #pragma unroll
    for (int r = 0; r < 8; ++r) {
      const int n = i0 + r + 8 * half;
      const int d = 16 * t + mn;
      ao[((size_t)(b * SEQ + n)) * EMB + h * HD + d] = acc[t][r];
    }
}

// ---------------------------------------------------------------------------
// Stage 4: out = AO[4096,2048] @ Wp[2048,2048] + b_proj
// ---------------------------------------------------------------------------
__global__ __launch_bounds__(32) void proj_gemm_kernel(
    const float* __restrict__ AO, const float* __restrict__ W,
    const float* __restrict__ bias, float* __restrict__ out) {
  const int lane = threadIdx.x & 31;
  const int half = lane >> 4;
  const int mn   = lane & 15;
  const int row0 = blockIdx.y * 32;
  const int col0 = blockIdx.x * 64;

  v8f acc[2][4];
  v8f z = {};
  #pragma unroll
  for (int i = 0; i < 2; ++i)
    #pragma unroll
    for (int t = 0; t < 4; ++t) acc[i][t] = z;

  const float* Ar0 = AO + (size_t)(row0 + mn) * EMB;
  const float* Ar1 = AO + (size_t)(row0 + 16 + mn) * EMB;

  for (int c = 0; c < EMB / 4; ++c) {
    const int kb = 4 * c + 2 * half;
    v2f a0, a1;
    a0.x = Ar0[kb]; a0.y = Ar0[kb + 1];
    a1.x = Ar1[kb]; a1.y = Ar1[kb + 1];
    const float* Wk0 = W + (size_t)kb * EMB + col0 + mn;
    const float* Wk1 = Wk0 + EMB;
    #pragma unroll
    for (int t = 0; t < 4; ++t) {
      v2f b; b.x = Wk0[16 * t]; b.y = Wk1[16 * t];
      acc[0][t] = wmma_f32(a0, b, acc[0][t]);
      acc[1][t] = wmma_f32(a1, b, acc[1][t]);
    }
  }

  #pragma unroll
  for (int mi = 0; mi < 2; ++mi)
    #pragma unroll
    for (int t = 0; t < 4; ++t)
      #pragma unroll
      for (int r = 0; r < 8; ++r) {
        const int gm = row0 + mi * 16 + r + 8 * half;
        const int gc = col0 + 16 * t + mn;
        out[(size_t)gm * EMB + gc] = acc[mi][t][r] + bias[gc];
      }
}

// ---------------------------------------------------------------------------
extern "C" void kernel_launch(void* const* d_in, const int* in_sizes, int n_in,
                              void* d_out, int out_size, void* d_ws,
                              size_t ws_size, hipStream_t stream) {
  (void)in_sizes; (void)n_in; (void)out_size; (void)ws_size;
  const float* X  = (const float*)d_in[0];  // [B,N,E]
  const float* Wf = (const float*)d_in[1];  // [E,3E]
  const float* Wp = (const float*)d_in[2];  // [E,E]
  const float* bp = (const float*)d_in[3];  // [E]

  float* out  = (float*)d_out;                        // [B,N,E]
  float* attn = out + (size_t)NB * SEQ * EMB;         // [B,H,N,N]

  const size_t headbuf = (size_t)BH * SEQ * HD;       // 8.4M floats each
  float* q  = (float*)d_ws;
  float* k  = q + headbuf;
  float* v  = k + headbuf;
  float* ao = q;  // Q is dead after attn_softmax_kernel; alias for attn_out

  dim3 blk(32);
  qkv_gemm_kernel<<<dim3(E3 / 64, ROWS / 32), blk, 0, stream>>>(X, Wf, q, k, v);
  attn_softmax_kernel<<<dim3(BH * NT16), blk, 16 * SEQ * sizeof(float), stream>>>(
      q, k, attn);
  pv_gemm_kernel<<<dim3(BH * NT16), blk, 0, stream>>>(attn, v, ao);
  proj_gemm_kernel<<<dim3(EMB / 64, ROWS / 32), blk, 0, stream>>>(ao, Wp, bp, out);
}